// VGClassifier_39367670235817
// MI455X (gfx1250) — compile-verified
//
#include <hip/hip_runtime.h>
#include <hip/hip_bf16.h>

typedef __attribute__((ext_vector_type(2))) float v2f;
typedef __attribute__((ext_vector_type(8))) float v8f;

#define BATCHES 8
#define SEQ     256
#define DIM     768

// ---------------------------------------------------------------------------
// Kernel 1: per-wave 16x16 fp32 Gram tile, G[b] = X[b] * X[b]^T, via
// V_WMMA_F32_16X16X4_F32 (true fp32 matrix op -> bit-compatible thresholds).
//
// A-matrix (16x4 f32) lane layout (ISA 7.12.2): lanes 0-15 hold M=lane with
// K=0 (v0), K=1 (v1); lanes 16-31 hold M=lane-16 with K=2 (v0), K=3 (v1).
// B-matrix (4x16) is the mirrored layout over N. For G = X*X^T both A and B
// tiles load with the identical pattern: row = base + (lane&15),
// cols = k + 2*(lane>>4) .. +1  -> one aligned float2 per matrix per K-step.
// C/D (16x16 f32, 8 VGPRs): row = rb + r + 8*(lane>>4), col = cb + (lane&15).
// ---------------------------------------------------------------------------
__global__ __launch_bounds__(256) void vvg_gram_wmma_f32(
    const float* __restrict__ X,   // [BATCHES, SEQ, DIM]
    float*       __restrict__ G)   // [BATCHES, SEQ, SEQ]
{
    const int lane  = threadIdx.x & 31;
    const int wave  = threadIdx.x >> 5;
    const int gwave = blockIdx.x * 8 + wave;          // 0 .. 2047
    const int batch = gwave >> 8;                     // / 256 tiles
    const int tile  = gwave & 255;
    const int rb    = (tile >> 4) << 4;               // tile row base
    const int cb    = (tile & 15) << 4;               // tile col base

    const float* Xb = X + batch * (SEQ * DIM);
    const int m  = lane & 15;
    const int hi = lane >> 4;                         // 0 or 1

    const float* arow = Xb + (rb + m) * DIM + 2 * hi; // A: rows rb..rb+15
    const float* brow = Xb + (cb + m) * DIM + 2 * hi; // B: rows cb..cb+15

    v8f acc = {};
#pragma unroll 4
    for (int k = 0; k < DIM; k += 4) {
        v2f a = *(const v2f*)(arow + k);
        v2f b = *(const v2f*)(brow + k);
        // 8 args: (neg_a, A, neg_b, B, c_mod, C, reuse_a, reuse_b)
        acc = __builtin_amdgcn_wmma_f32_16x16x4_f32(
            false, a, false, b, (short)0, acc, false, false);
    }

    float* Gb = G + batch * (SEQ * SEQ);
    const int col     = cb + m;
    const int rowbase = rb + 8 * hi;
#pragma unroll
    for (int r = 0; r < 8; ++r) {
        Gb[(rowbase + r) * SEQ + col] = acc[r];
    }
}

// ---------------------------------------------------------------------------
// Kernel 2: one block per batch.
//  Phase A: thread b decides visibility of node 0 -> node b (normalization by
//           ||x_b|| cancels from the reference predicate, so raw G works).
//  Phase B: h = sum_b vis[b]*x_b, partial logits = h . W, LDS tree-reduce,
//           log_softmax, write 4 floats.
// ---------------------------------------------------------------------------
__global__ __launch_bounds__(256) void vvg_classify(
    const float* __restrict__ X,   // [BATCHES, SEQ, DIM]
    const float* __restrict__ G,   // [BATCHES, SEQ, SEQ]
    const float* __restrict__ W,   // [DIM, 4]
    float*       __restrict__ out) // [BATCHES, 4]
{
    const int batch = blockIdx.x;
    const int t     = threadIdx.x;

    __shared__ float vis[SEQ];
    __shared__ float red[SEQ * 4];

    const float* Gb = G + batch * (SEQ * SEQ);
    const float* Xb = X + batch * (SEQ * DIM);

    // ---- Phase A: visibility of node 0 to node t -------------------------
    float v = 0.0f;                       // node 0 itself: row0[0] = 0
    if (t >= 1) {
        const float* grow = Gb + t * SEQ; // grow[c] = <x_t, x_c>
        const float  Gbb  = grow[t];      // <x_b, x_b>
        const float  G0   = grow[0];      // <x_0, x_b>
        const float  invb = 1.0f / (float)t;
        bool blocked = false;
        for (int c = 1; c < t; ++c) {
            float line = Gbb + (G0 - Gbb) * ((float)(t - c) * invb);
            blocked = blocked || (grow[c] >= line);
        }
        v = blocked ? 0.0f : 1.0f;
    }
    vis[t] = v;
    __syncthreads();

    // ---- Phase B: h = vis . X ; logits = h . W ---------------------------
    float p0 = 0.f, p1 = 0.f, p2 = 0.f, p3 = 0.f;
    for (int d = t; d < DIM; d += 256) {
        float hd = 0.0f;
        for (int b = 0; b < SEQ; ++b) {
            hd = fmaf(vis[b], Xb[b * DIM + d], hd);
        }
        p0 = fmaf(hd, W[d * 4 + 0], p0);
        p1 = fmaf(hd, W[d * 4 + 1], p1);
        p2 = fmaf(hd, W[d * 4 + 2], p2);
        p3 = fmaf(hd, W[d * 4 + 3], p3);
    }
    red[t * 4 + 0] = p0;
    red[t * 4 + 1] = p1;
    red[t * 4 + 2] = p2;
    red[t * 4 + 3] = p3;
    __syncthreads();

    for (int stride = 128; stride >= 1; stride >>= 1) {
        if (t < stride) {
            red[t * 4 + 0] += red[(t + stride) * 4 + 0];
            red[t * 4 + 1] += red[(t + stride) * 4 + 1];
            red[t * 4 + 2] += red[(t + stride) * 4 + 2];
            red[t * 4 + 3] += red[(t + stride) * 4 + 3];
        }
        __syncthreads();
    }

    if (t == 0) {
        float l0 = red[0], l1 = red[1], l2 = red[2], l3 = red[3];
        float mx  = fmaxf(fmaxf(l0, l1), fmaxf(l2, l3));
        float se  = __expf(l0 - mx) + __expf(l1 - mx) +
                    __expf(l2 - mx) + __expf(l3 - mx);
        float lse = mx + __logf(se);
        out[batch * 4 + 0] = l0 - lse;
        out[batch * 4 + 1] = l1 - lse;
        out[batch * 4 + 2] = l2 - lse;
        out[batch * 4 + 3] = l3 - lse;
    }
}

extern "C" void kernel_launch(void* const* d_in, const int* in_sizes, int n_in,
                              void* d_out, int out_size, void* d_ws, size_t ws_size,
                              hipStream_t stream) {
    (void)in_sizes; (void)n_in; (void)out_size; (void)ws_size;
    const float* X = (const float*)d_in[0];   // [8, 256, 768] fp32
    const float* W = (const float*)d_in[1];   // [768, 4] fp32
    float* out = (float*)d_out;               // [8, 4] fp32
    float* G   = (float*)d_ws;                // needs 8*256*256*4 = 2 MB

    // 8 batches * 256 tiles, 8 waves (256 thr) per block -> 256 blocks
    vvg_gram_wmma_f32<<<256, 256, 0, stream>>>(X, G);
    vvg_classify<<<BATCHES, 256, 0, stream>>>(X, G, W, out);
}